// SequentialVAE_29970281791617
// MI455X (gfx1250) — compile-verified
//
#include <hip/hip_runtime.h>
#include <math.h>

typedef __attribute__((ext_vector_type(2))) float v2f;
typedef __attribute__((ext_vector_type(8))) float v8f;

#define T_  512
#define B_  64
#define D_  1024
#define LZ_ 9
#define LF_ 1296
#define KC_ 1027   // D + H
#define KP_ 1028   // padded K for out_f GEMM
#define NO_ 2592   // 2*LF

// ---- output offsets (floats) in d_out, in reference return order ----
#define O_MUF (T_*B_*D_)
#define O_LVF (O_MUF + B_*LF_)
#define O_MUP (O_LVF + B_*LF_)
#define O_LVP (O_MUP + T_*B_*LZ_)
#define O_MUZ (O_LVP + T_*B_*LZ_)
#define O_LVZ (O_MUZ + T_*B_*LZ_)

// ---- workspace offsets (floats) ----
#define W_WCAT 0                       // 16 x 1024 packed [fc1_w ; rnn_Wx ; 0]
#define W_H1   (W_WCAT + 16*1024)      // T*B*3  relu(x@fc1^T+b)
#define W_U    (W_H1 + T_*B_*3)        // T*B*3  x@Wx^T + b_rnn
#define W_HL   (W_U + T_*B_*3)         // 64*3   h_last
#define W_COMB (W_HL + 192)            // 64*1028 padded comb_last
#define W_F    (W_COMB + 64*KP_)       // 64*1296 sampled f
#define W_G    (W_F + B_*LF_)          // 64*3    f @ fc3[:,9:]^T
#define W_Z    (W_G + 192)             // T*B*9   sampled z
#define W_C    (W_Z + T_*B_*LZ_)       // T*B*3   prior c_t = Wzh z_t + b
#define W_DCH  (W_C + T_*B_*3)         // 16*64*3 chunk partials
#define W_HB   (W_DCH + 16*64*3)       // 16*64*3 chunk boundary states

// ---------------- deterministic hash-normal ----------------
__device__ __forceinline__ unsigned pcg_(unsigned v) {
  v = v * 747796405u + 2891336453u;
  unsigned w = ((v >> ((v >> 28) + 4u)) ^ v) * 277803737u;
  return (w >> 22) ^ w;
}
__device__ __forceinline__ float rnorm_(unsigned idx, unsigned salt) {
  unsigned a = pcg_(idx * 2u + salt * 0x9E3779B9u);
  unsigned b = pcg_(idx * 2u + 1u + salt * 0x85EBCA6Bu);
  float u1 = (float)(a | 1u) * 2.3283064365386963e-10f;
  float u2 = (float)b * 2.3283064365386963e-10f;
  float r = sqrtf(-2.0f * __logf(fmaxf(u1, 1e-12f)));
  return r * __cosf(6.28318530717958647f * u2);
}

// ---------------- K0: pack [fc1_w ; rnn_Wx ; zeros] into 16x1024 ----------------
__global__ void k_pack(const float* __restrict__ fc1_w,
                       const float* __restrict__ rnn_i2h_w,
                       float* __restrict__ ws) {
  int i = blockIdx.x * blockDim.x + threadIdx.x;
  if (i >= 16 * 1024) return;
  int n = i >> 10, k = i & 1023;
  float v = 0.f;
  if (n < 3)      v = fc1_w[n * 1024 + k];
  else if (n < 6) v = rnn_i2h_w[(n - 3) * KC_ + k];
  ws[W_WCAT + n * 1024 + k] = v;
}

// ---------------- K1: fused pass over x : [h1 | u] = x @ Wcat^T  (WMMA f32 16x16x4) ----
__global__ void __launch_bounds__(32) k_xpass(const float* __restrict__ x,
                                              const float* __restrict__ fc1_b,
                                              const float* __restrict__ rnn_b,
                                              float* __restrict__ ws) {
  int rowbase = blockIdx.x * 16;            // tile over T*B rows
  int lane = threadIdx.x;
  int half = lane >> 4, l15 = lane & 15;
  const float* arow = x + (size_t)(rowbase + l15) * D_;
  const float* brow = ws + W_WCAT + l15 * 1024;
  v8f c = {};
  for (int kc = 0; kc < D_; kc += 4) {
    int k0 = kc + half * 2;
    v2f a = *(const v2f*)(arow + k0);       // A: 16x4 f32, M=l15, K=half*2+{0,1}
    v2f b = *(const v2f*)(brow + k0);       // B: 4x16 f32, N=l15, same K pair
    c = __builtin_amdgcn_wmma_f32_16x16x4_f32(false, a, false, b, (short)0, c,
                                              false, false);
  }
#pragma unroll
  for (int i = 0; i < 8; ++i) {             // C: vgpr i -> M = i + 8*half, N = l15
    int m = rowbase + i + half * 8;
    int n = l15;
    float v = c[i];
    if (n < 3)      ws[W_H1 + m * 3 + n] = fmaxf(v + fc1_b[n], 0.f);
    else if (n < 6) ws[W_U + m * 3 + (n - 3)] = v + rnn_b[n - 3];
  }
}

// ---------------- K2: 3-wide linear scan for h_last (511 steps, 64 lanes) ----------
__global__ void k_frnn(const float* __restrict__ rnn_i2h_w, float* __restrict__ ws) {
  int b = threadIdx.x;
  if (b >= 64) return;
  float w00 = rnn_i2h_w[0 * KC_ + 1024], w01 = rnn_i2h_w[0 * KC_ + 1025], w02 = rnn_i2h_w[0 * KC_ + 1026];
  float w10 = rnn_i2h_w[1 * KC_ + 1024], w11 = rnn_i2h_w[1 * KC_ + 1025], w12 = rnn_i2h_w[1 * KC_ + 1026];
  float w20 = rnn_i2h_w[2 * KC_ + 1024], w21 = rnn_i2h_w[2 * KC_ + 1025], w22 = rnn_i2h_w[2 * KC_ + 1026];
  float h0 = 0.f, h1 = 0.f, h2 = 0.f;
  for (int t = 0; t < T_ - 1; ++t) {
    const float* u = ws + W_U + (t * B_ + b) * 3;
    float u0 = u[0], u1 = u[1], u2 = u[2];
    float n0 = fmaf(w00, h0, fmaf(w01, h1, fmaf(w02, h2, u0)));
    float n1 = fmaf(w10, h0, fmaf(w11, h1, fmaf(w12, h2, u1)));
    float n2 = fmaf(w20, h0, fmaf(w21, h1, fmaf(w22, h2, u2)));
    h0 = n0; h1 = n1; h2 = n2;
  }
  ws[W_HL + b * 3 + 0] = h0; ws[W_HL + b * 3 + 1] = h1; ws[W_HL + b * 3 + 2] = h2;
}

// ---------------- K2b: materialize padded comb_last (64 x 1028) ----------------
__global__ void k_comb(const float* __restrict__ x, float* __restrict__ ws) {
  int i = blockIdx.x * blockDim.x + threadIdx.x;
  if (i >= 64 * KP_) return;
  int m = i / KP_, k = i % KP_;
  float v = 0.f;
  if (k < 1024)     v = x[(size_t)((T_ - 1) * B_ + m) * D_ + k];
  else if (k < KC_) v = ws[W_HL + m * 3 + (k - 1024)];
  ws[W_COMB + m * KP_ + k] = v;
}

// ---------------- K3: out_f = comb_last @ rnn_i2o_w^T + b  (WMMA, 4x162 tiles) -----
__global__ void __launch_bounds__(32) k_outf(const float* __restrict__ W,
                                             const float* __restrict__ bias,
                                             const float* __restrict__ ws,
                                             float* __restrict__ out) {
  int ntile = blockIdx.x, mtile = blockIdx.y;
  int lane = threadIdx.x, half = lane >> 4, l15 = lane & 15;
  const float* arow = ws + W_COMB + (mtile * 16 + l15) * KP_;
  const float* brow = W + (size_t)(ntile * 16 + l15) * KC_;   // row of rnn_i2o_w
  v8f c = {};
  for (int kc = 0; kc < 1024; kc += 4) {
    int k0 = kc + half * 2;
    v2f a = *(const v2f*)(arow + k0);
    v2f b; b.x = brow[k0]; b.y = brow[k0 + 1];   // stride 1027: scalar (4B-aligned)
    c = __builtin_amdgcn_wmma_f32_16x16x4_f32(false, a, false, b, (short)0, c,
                                              false, false);
  }
  { // tail chunk K = 1024..1027 (padded)
    int k0 = 1024 + half * 2;
    v2f a = *(const v2f*)(arow + k0);
    v2f b; b.x = (k0 < KC_) ? brow[k0] : 0.f;
    b.y = (k0 + 1 < KC_) ? brow[k0 + 1] : 0.f;
    c = __builtin_amdgcn_wmma_f32_16x16x4_f32(false, a, false, b, (short)0, c,
                                              false, false);
  }
#pragma unroll
  for (int i = 0; i < 8; ++i) {
    int m = mtile * 16 + i + half * 8;
    int col = ntile * 16 + l15;
    float v = c[i] + bias[col];
    if (col < LF_) out[O_MUF + m * LF_ + col] = v;
    else           out[O_LVF + m * LF_ + (col - LF_)] = v;
  }
}

// ---------------- K4: sample f ----------------
__global__ void k_fsample(const float* __restrict__ out, float* __restrict__ ws) {
  int i = blockIdx.x * blockDim.x + threadIdx.x;
  if (i >= B_ * LF_) return;
  float mu = out[O_MUF + i], lv = out[O_LVF + i];
  ws[W_F + i] = mu + rnorm_((unsigned)i, 1u) * __expf(0.5f * lv);
}

// ---------------- K5: z path + prior parallel parts (per (t,b)) ----------------
__global__ void k_zpass(const float* __restrict__ fc21_w, const float* __restrict__ fc21_b,
                        const float* __restrict__ fc22_w, const float* __restrict__ fc22_b,
                        const float* __restrict__ pr_i2h_w, const float* __restrict__ pr_i2h_b,
                        const float* __restrict__ pr_i2o_w, const float* __restrict__ pr_i2o_b,
                        float* __restrict__ ws, float* __restrict__ out) {
  int r = blockIdx.x * blockDim.x + threadIdx.x;
  if (r >= T_ * B_) return;
  float a0 = ws[W_H1 + r * 3 + 0], a1 = ws[W_H1 + r * 3 + 1], a2 = ws[W_H1 + r * 3 + 2];
  float zl[9];
#pragma unroll
  for (int j = 0; j < 9; ++j) {
    float mu = fmaf(fc21_w[j*3], a0, fmaf(fc21_w[j*3+1], a1, fmaf(fc21_w[j*3+2], a2, fc21_b[j])));
    float lv = fmaf(fc22_w[j*3], a0, fmaf(fc22_w[j*3+1], a1, fmaf(fc22_w[j*3+2], a2, fc22_b[j])));
    float zv = mu + rnorm_((unsigned)(r * 9 + j), 2u) * __expf(0.5f * lv);
    out[O_MUZ + r * 9 + j] = mu;
    out[O_LVZ + r * 9 + j] = lv;
    ws[W_Z + r * 9 + j] = zv;
    zl[j] = zv;
  }
#pragma unroll
  for (int j = 0; j < 3; ++j) {          // c_t = Wzh z + b  (prior hidden drive)
    float s = pr_i2h_b[j];
#pragma unroll
    for (int k = 0; k < 9; ++k) s = fmaf(pr_i2h_w[j * 12 + k], zl[k], s);
    ws[W_C + r * 3 + j] = s;
  }
#pragma unroll
  for (int o = 0; o < 18; ++o) {         // p_t = Uz z + b_o  (h-independent part)
    float s = pr_i2o_b[o];
#pragma unroll
    for (int k = 0; k < 9; ++k) s = fmaf(pr_i2o_w[o * 12 + k], zl[k], s);
    if (o < 9) out[O_MUP + r * 9 + o] = s;
    else       out[O_LVP + r * 9 + (o - 9)] = s;
  }
}

// ---------------- K6: g = f @ fc3[:,9:]^T (64 x 3) ----------------
__global__ void k_g(const float* __restrict__ fc3_w, float* __restrict__ ws) {
  int i = blockIdx.x * blockDim.x + threadIdx.x;
  if (i >= 192) return;
  int b = i / 3, j = i % 3;
  const float* fr = ws + W_F + b * LF_;
  const float* wr = fc3_w + j * 1305 + 9;
  float s = 0.f;
  for (int k = 0; k < LF_; ++k) s = fmaf(fr[k], wr[k], s);
  ws[W_G + b * 3 + j] = s;
}

// ---------------- K7: recon_x = relu(z@Wz3^T + g + b3) @ fc4_w^T + b4 --------------
__global__ void __launch_bounds__(256) k_recon(const float* __restrict__ fc3_w,
                                               const float* __restrict__ fc3_b,
                                               const float* __restrict__ fc4_w,
                                               const float* __restrict__ fc4_b,
                                               const float* __restrict__ ws,
                                               float* __restrict__ out) {
  int row = blockIdx.x;                  // (t*64 + b)
  int b = row & 63;
  const float* zr = ws + W_Z + row * 9;
  float h3[3];
#pragma unroll
  for (int h = 0; h < 3; ++h) {
    float s = fc3_b[h] + ws[W_G + b * 3 + h];
    const float* wr = fc3_w + h * 1305;
#pragma unroll
    for (int j = 0; j < 9; ++j) s = fmaf(wr[j], zr[j], s);
    h3[h] = fmaxf(s, 0.f);
  }
  int d0 = threadIdx.x * 4;
  float r[4];
#pragma unroll
  for (int q = 0; q < 4; ++q) {
    int d = d0 + q;
    r[q] = fmaf(h3[0], fc4_w[d * 3 + 0],
            fmaf(h3[1], fc4_w[d * 3 + 1],
             fmaf(h3[2], fc4_w[d * 3 + 2], fc4_b[d])));
  }
  float4 o4; o4.x = r[0]; o4.y = r[1]; o4.z = r[2]; o4.w = r[3];
  *(float4*)(out + (size_t)row * D_ + d0) = o4;
}

// ---------------- K8a: prior scan chunk partials d_j (16 chunks x 64 batch) --------
__global__ void k_pchunk(const float* __restrict__ pr_i2h_w, float* __restrict__ ws) {
  int i = blockIdx.x * blockDim.x + threadIdx.x;
  if (i >= 1024) return;
  int ch = i >> 6, b = i & 63;
  float A[9];
#pragma unroll
  for (int j = 0; j < 3; ++j)
#pragma unroll
    for (int k = 0; k < 3; ++k) A[j * 3 + k] = pr_i2h_w[j * 12 + 9 + k];
  float d0 = 0.f, d1 = 0.f, d2 = 0.f;
  for (int s = 0; s < 32; ++s) {
    const float* cc = ws + W_C + ((ch * 32 + s) * B_ + b) * 3;
    float n0 = fmaf(A[0], d0, fmaf(A[1], d1, fmaf(A[2], d2, cc[0])));
    float n1 = fmaf(A[3], d0, fmaf(A[4], d1, fmaf(A[5], d2, cc[1])));
    float n2 = fmaf(A[6], d0, fmaf(A[7], d1, fmaf(A[8], d2, cc[2])));
    d0 = n0; d1 = n1; d2 = n2;
  }
  ws[W_DCH + (ch * 64 + b) * 3 + 0] = d0;
  ws[W_DCH + (ch * 64 + b) * 3 + 1] = d1;
  ws[W_DCH + (ch * 64 + b) * 3 + 2] = d2;
}

// ---------------- K8b: 16-step boundary scan (h at each chunk start) ---------------
__global__ void k_pbound(const float* __restrict__ pr_i2h_w, float* __restrict__ ws) {
  int b = threadIdx.x;
  if (b >= 64) return;
  float M[9];
#pragma unroll
  for (int j = 0; j < 3; ++j)
#pragma unroll
    for (int k = 0; k < 3; ++k) M[j * 3 + k] = pr_i2h_w[j * 12 + 9 + k];
  // M = A^32 by 5 squarings
  for (int q = 0; q < 5; ++q) {
    float N[9];
#pragma unroll
    for (int r = 0; r < 3; ++r)
#pragma unroll
      for (int c = 0; c < 3; ++c)
        N[r * 3 + c] = fmaf(M[r * 3], M[c],
                        fmaf(M[r * 3 + 1], M[3 + c], M[r * 3 + 2] * M[6 + c]));
#pragma unroll
    for (int t = 0; t < 9; ++t) M[t] = N[t];
  }
  float h0 = 0.f, h1 = 0.f, h2 = 0.f;
  for (int j = 0; j < 16; ++j) {
    ws[W_HB + (j * 64 + b) * 3 + 0] = h0;
    ws[W_HB + (j * 64 + b) * 3 + 1] = h1;
    ws[W_HB + (j * 64 + b) * 3 + 2] = h2;
    const float* dd = ws + W_DCH + (j * 64 + b) * 3;
    float n0 = fmaf(M[0], h0, fmaf(M[1], h1, fmaf(M[2], h2, dd[0])));
    float n1 = fmaf(M[3], h0, fmaf(M[4], h1, fmaf(M[5], h2, dd[1])));
    float n2 = fmaf(M[6], h0, fmaf(M[7], h1, fmaf(M[8], h2, dd[2])));
    h0 = n0; h1 = n1; h2 = n2;
  }
}

// ---------------- K8c: replay chunks, add Uh*h_t into prior outputs ----------------
__global__ void k_pout(const float* __restrict__ pr_i2h_w,
                       const float* __restrict__ pr_i2o_w,
                       const float* __restrict__ ws, float* __restrict__ out) {
  int i = blockIdx.x * blockDim.x + threadIdx.x;
  if (i >= 1024) return;
  int ch = i >> 6, b = i & 63;
  float A[9], U[54];
#pragma unroll
  for (int j = 0; j < 3; ++j)
#pragma unroll
    for (int k = 0; k < 3; ++k) A[j * 3 + k] = pr_i2h_w[j * 12 + 9 + k];
#pragma unroll
  for (int o = 0; o < 18; ++o)
#pragma unroll
    for (int k = 0; k < 3; ++k) U[o * 3 + k] = pr_i2o_w[o * 12 + 9 + k];
  float h0 = ws[W_HB + (ch * 64 + b) * 3 + 0];
  float h1 = ws[W_HB + (ch * 64 + b) * 3 + 1];
  float h2 = ws[W_HB + (ch * 64 + b) * 3 + 2];
  for (int s = 0; s < 32; ++s) {
    int r = (ch * 32 + s) * B_ + b;
#pragma unroll
    for (int o = 0; o < 9; ++o) {
      float am = fmaf(U[o * 3], h0, fmaf(U[o * 3 + 1], h1, U[o * 3 + 2] * h2));
      float av = fmaf(U[(o + 9) * 3], h0,
                  fmaf(U[(o + 9) * 3 + 1], h1, U[(o + 9) * 3 + 2] * h2));
      out[O_MUP + r * 9 + o] = out[O_MUP + r * 9 + o] + am;
      out[O_LVP + r * 9 + o] = out[O_LVP + r * 9 + o] + av;
    }
    const float* cc = ws + W_C + r * 3;
    float n0 = fmaf(A[0], h0, fmaf(A[1], h1, fmaf(A[2], h2, cc[0])));
    float n1 = fmaf(A[3], h0, fmaf(A[4], h1, fmaf(A[5], h2, cc[1])));
    float n2 = fmaf(A[6], h0, fmaf(A[7], h1, fmaf(A[8], h2, cc[2])));
    h0 = n0; h1 = n1; h2 = n2;
  }
}

extern "C" void kernel_launch(void* const* d_in, const int* in_sizes, int n_in,
                              void* d_out, int out_size, void* d_ws, size_t ws_size,
                              hipStream_t stream) {
  const float* x         = (const float*)d_in[0];
  const float* fc1_w     = (const float*)d_in[1];
  const float* fc1_b     = (const float*)d_in[2];
  const float* fc21_w    = (const float*)d_in[3];
  const float* fc21_b    = (const float*)d_in[4];
  const float* fc22_w    = (const float*)d_in[5];
  const float* fc22_b    = (const float*)d_in[6];
  const float* rnn_i2h_w = (const float*)d_in[7];
  const float* rnn_i2h_b = (const float*)d_in[8];
  const float* rnn_i2o_w = (const float*)d_in[9];
  const float* rnn_i2o_b = (const float*)d_in[10];
  const float* fc3_w     = (const float*)d_in[11];
  const float* fc3_b     = (const float*)d_in[12];
  const float* fc4_w     = (const float*)d_in[13];
  const float* fc4_b     = (const float*)d_in[14];
  const float* pr_i2h_w  = (const float*)d_in[15];
  const float* pr_i2h_b  = (const float*)d_in[16];
  const float* pr_i2o_w  = (const float*)d_in[17];
  const float* pr_i2o_b  = (const float*)d_in[18];
  float* out = (float*)d_out;
  float* ws  = (float*)d_ws;

  k_pack<<<(16 * 1024 + 255) / 256, 256, 0, stream>>>(fc1_w, rnn_i2h_w, ws);
  k_xpass<<<(T_ * B_) / 16, 32, 0, stream>>>(x, fc1_b, rnn_i2h_b, ws);
  k_frnn<<<1, 64, 0, stream>>>(rnn_i2h_w, ws);
  k_comb<<<(64 * KP_ + 255) / 256, 256, 0, stream>>>(x, ws);
  k_outf<<<dim3(NO_ / 16, B_ / 16), 32, 0, stream>>>(rnn_i2o_w, rnn_i2o_b, ws, out);
  k_fsample<<<(B_ * LF_ + 255) / 256, 256, 0, stream>>>(out, ws);
  k_zpass<<<(T_ * B_) / 256, 256, 0, stream>>>(fc21_w, fc21_b, fc22_w, fc22_b,
                                               pr_i2h_w, pr_i2h_b, pr_i2o_w,
                                               pr_i2o_b, ws, out);
  k_g<<<1, 192, 0, stream>>>(fc3_w, ws);
  k_recon<<<T_ * B_, 256, 0, stream>>>(fc3_w, fc3_b, fc4_w, fc4_b, ws, out);
  k_pchunk<<<4, 256, 0, stream>>>(pr_i2h_w, ws);
  k_pbound<<<1, 64, 0, stream>>>(pr_i2h_w, ws);
  k_pout<<<4, 256, 0, stream>>>(pr_i2h_w, pr_i2o_w, ws, out);
}